// QEqModule_3745211483115
// MI455X (gfx1250) — compile-verified
//
#include <hip/hip_runtime.h>
#include <stdint.h>

// ---------------------------------------------------------------------------
// QEq Coulomb + Ewald-real-space edge kernel for MI455X (gfx1250).
// Strategy: stream 20B/edge (dij[3]f32 + row i32 + col i32) through LDS using
// the CDNA5 async global->LDS engine (ASYNCcnt), double-buffered per wave,
// scatter with agent-scope f32 atomics (resolve in 192MB L2).
// Units refactored to Angstrom/eV so all f32 intermediates are O(1..100).
// ---------------------------------------------------------------------------

#define CHUNK_EDGES   128                  // per-wave chunk
#define CHUNK_BYTES   (CHUNK_EDGES * 20)   // 2560B: dij 1536 | row 512 | col 512
#define WAVES_PER_BLK 8
#define BLK_THREADS   256
#define EDGE_BLOCKS   1024

#define QQR2E   14.399645478f   // k_e * e * 1e10  ->  eV*Angstrom
#define R0_A    2.2f
#define BETA    18.7f
#define B_OVR0  (18.7f / 2.2f)
#define EW_P    0.3275911f
#define EW_A0   0.254829592f
#define EW_A1  -0.284496736f
#define EW_A2   1.421413741f
#define EW_A3  -1.453152027f
#define EW_A4   1.061405429f

__device__ __forceinline__ void atomic_fadd(float* p, float v) {
  // relaxed, agent scope -> global_atomic_add_f32 (no return, STOREcnt)
  __hip_atomic_fetch_add(p, v, __ATOMIC_RELAXED, __HIP_MEMORY_SCOPE_AGENT);
}

// CDNA5 async copy: 16B global -> LDS, tracked by ASYNCcnt (per-wave counter).
__device__ __forceinline__ void async_b128(unsigned lds_addr, const void* gptr) {
  unsigned long long ga = (unsigned long long)(uintptr_t)gptr;
  asm volatile("global_load_async_to_lds_b128 %0, %1, off"
               :: "v"(lds_addr), "v"(ga) : "memory");
}
__device__ __forceinline__ void wait_async_le5() {
  asm volatile("s_wait_asynccnt 0x5" ::: "memory");
}
__device__ __forceinline__ void wait_async_le0() {
  asm volatile("s_wait_asynccnt 0x0" ::: "memory");
}

// Issue one chunk: exactly 5 wave-uniform async b128 instructions (full EXEC),
// so ASYNCcnt advances by exactly 5 per chunk per wave.
__device__ __forceinline__ void issue_chunk(const char* dij_b, const char* row_b,
                                            const char* col_b, long long chunk,
                                            unsigned lds_base, int lane) {
  const char* gd = dij_b + chunk * (long long)(CHUNK_EDGES * 12);
  const char* gr = row_b + chunk * (long long)(CHUNK_EDGES * 4);
  const char* gc = col_b + chunk * (long long)(CHUNK_EDGES * 4);
#pragma unroll
  for (int i = 0; i < 3; ++i) {   // dij: 1536B = 3 * 32 lanes * 16B
    unsigned off = (unsigned)((i * 32 + lane) * 16);
    async_b128(lds_base + off, gd + off);
  }
  async_b128(lds_base + 1536u + (unsigned)(lane * 16), gr + lane * 16);  // row
  async_b128(lds_base + 2048u + (unsigned)(lane * 16), gc + lane * 16);  // col
}

__device__ __forceinline__ void edge_compute(float dx, float dy, float dz,
                                             int r, int c, const float* q,
                                             float gA, float* e_atom, float* F) {
  float r2   = dx * dx + dy * dy + dz * dz;
  float rinv = __builtin_amdgcn_rsqf(r2);     // v_rsq_f32
  float rA   = r2 * rinv;                     // |d| in Angstrom
  float cc   = q[r] * q[c];
  float pref = QQR2E * cc * rinv;             // eV
  float damp = 1.0f;
  if (rA < R0_A) damp = __expf(fmaf(rA, B_OVR0, -BETA));
  float grij  = gA * rA;                      // dimensionless
  float expm2 = __expf(-grij * grij);
  float t     = __builtin_amdgcn_rcpf(fmaf(EW_P, grij, 1.0f));  // v_rcp_f32
  float erfc  = t * fmaf(t, fmaf(t, fmaf(t, fmaf(t, EW_A4, EW_A3), EW_A2), EW_A1), EW_A0) * expm2;
  float ec = pref * (0.5f * damp + erfc - 1.0f);                       // eV
  float fs = pref * (damp + erfc + 2.0f * grij * expm2 - 1.0f) * (rinv * rinv); // eV/A^2
  atomic_fadd(e_atom + r, ec);
  float fx = dx * fs, fy = dy * fs, fz = dz * fs;
  atomic_fadd(F + 3 * r + 0,  fx);
  atomic_fadd(F + 3 * r + 1,  fy);
  atomic_fadd(F + 3 * r + 2,  fz);
  atomic_fadd(F + 3 * c + 0, -fx);
  atomic_fadd(F + 3 * c + 1, -fy);
  atomic_fadd(F + 3 * c + 2, -fz);
}

__global__ __launch_bounds__(BLK_THREADS)
void qeq_edge_kernel(const float* __restrict__ dij, const float* __restrict__ q,
                     const float* __restrict__ gsrc, const int* __restrict__ row,
                     const int* __restrict__ col, float* __restrict__ e_atom,
                     float* __restrict__ F, long long E, int nchunks) {
  __shared__ __align__(16) char lds[WAVES_PER_BLK * 2 * CHUNK_BYTES];

  const int   lane   = threadIdx.x & 31;
  const int   gwave  = (blockIdx.x * BLK_THREADS + threadIdx.x) >> 5;
  const int   nwaves = (gridDim.x * BLK_THREADS) >> 5;
  const float gA     = gsrc[0] * 1e-10f;     // g_ewald in 1/Angstrom

  const unsigned lds0 = (unsigned)(uintptr_t)lds +
                        (unsigned)((threadIdx.x >> 5) * 2 * CHUNK_BYTES);
  const char* dij_b = (const char*)dij;
  const char* row_b = (const char*)row;
  const char* col_b = (const char*)col;

  long long c = gwave;
  int buf = 0;
  if (c < nchunks) issue_chunk(dij_b, row_b, col_b, c, lds0, lane);

  for (; c < nchunks; c += nwaves) {
    long long cn = c + nwaves;
    if (cn < nchunks) {                      // prefetch next chunk, keep 5 in flight
      issue_chunk(dij_b, row_b, col_b, cn, lds0 + (unsigned)((buf ^ 1) * CHUNK_BYTES), lane);
      wait_async_le5();                      // current buffer's 5 ops are done
    } else {
      wait_async_le0();
    }
    const char*  base = lds + (size_t)(lds0 - (unsigned)(uintptr_t)lds) + (size_t)buf * CHUNK_BYTES;
    const float* sdij = (const float*)base;
    const int*   srow = (const int*)(base + 1536);
    const int*   scol = (const int*)(base + 2048);
#pragma unroll
    for (int k = 0; k < 4; ++k) {
      int j = lane + 32 * k;
      float dx = sdij[3 * j + 0];
      float dy = sdij[3 * j + 1];
      float dz = sdij[3 * j + 2];
      edge_compute(dx, dy, dz, srow[j], scol[j], q, gA, e_atom, F);
    }
    buf ^= 1;
  }

  // tail edges (< CHUNK_EDGES), direct loads
  long long tail0 = (long long)nchunks * CHUNK_EDGES;
  long long e = tail0 + (long long)(blockIdx.x * BLK_THREADS + threadIdx.x);
  if (e < E) {
    size_t i3 = (size_t)e * 3;
    edge_compute(dij[i3], dij[i3 + 1], dij[i3 + 2], row[e], col[e], q, gA, e_atom, F);
  }
}

__global__ void qeq_reduce_kernel(const float* __restrict__ e_atom,
                                  const int* __restrict__ batch,
                                  float* __restrict__ energy, int n_atoms) {
  int i = blockIdx.x * blockDim.x + threadIdx.x;
  if (i < n_atoms) atomic_fadd(energy + batch[i], e_atom[i]);
}

__global__ void qeq_zero_kernel(float* __restrict__ out, long long n_out,
                                float* __restrict__ ws, long long n_ws) {
  long long stride = (long long)gridDim.x * blockDim.x;
  for (long long i = blockIdx.x * (long long)blockDim.x + threadIdx.x; i < n_out; i += stride)
    out[i] = 0.0f;
  for (long long i = blockIdx.x * (long long)blockDim.x + threadIdx.x; i < n_ws; i += stride)
    ws[i] = 0.0f;
}

extern "C" void kernel_launch(void* const* d_in, const int* in_sizes, int n_in,
                              void* d_out, int out_size, void* d_ws, size_t ws_size,
                              hipStream_t stream) {
  const float* dij   = (const float*)d_in[0];
  const float* q     = (const float*)d_in[1];
  const float* gsrc  = (const float*)d_in[2];
  const int*   row   = (const int*)d_in[3];
  const int*   col   = (const int*)d_in[4];
  const int*   batch = (const int*)d_in[5];

  const long long E = in_sizes[3];
  const int N       = in_sizes[1];
  const int G       = out_size - 3 * N;     // 256 graph energies

  float* energy = (float*)d_out;            // [G]
  float* force  = (float*)d_out + G;        // [N*3]
  float* e_atom = (float*)d_ws;             // [N] per-atom energy scratch

  qeq_zero_kernel<<<512, 256, 0, stream>>>(energy, (long long)out_size,
                                           e_atom, (long long)N);

  int nchunks = (int)(E / CHUNK_EDGES);
  qeq_edge_kernel<<<EDGE_BLOCKS, BLK_THREADS, 0, stream>>>(
      dij, q, gsrc, row, col, e_atom, force, E, nchunks);

  qeq_reduce_kernel<<<(N + 255) / 256, 256, 0, stream>>>(e_atom, batch, energy, N);
}